// ContradictionDetector_47502338294426
// MI455X (gfx1250) — compile-verified
//
#include <hip/hip_runtime.h>
#include <hip/hip_bf16.h>
#include <math.h>

// Problem constants (from the reference)
#define NN 50000
#define DD 128
#define HH 256
#define EE 400000
#define KK 256            // 2*D contraction depth
#define MT 32             // edges / active-nodes per block (two 16-row WMMA subtiles)
#define PSTR 260          // padded LDS row stride (floats): rows land on distinct banks

typedef float v2f __attribute__((ext_vector_type(2)));
typedef float v8f __attribute__((ext_vector_type(8)));

// ---------------------------------------------------------------------------
// WMMA helper: D = A(16x4 f32) * B(4x16 f32) + C, full f32 (V_WMMA_F32_16X16X4_F32)
// A layout: lane 0-15 -> M=lane, VGPR0/1 = K=0/1 ; lane 16-31 -> K=2/3
// B layout: VGPR v, lanes 0-15 -> K=v, lanes 16-31 -> K=v+2, N=lane&15
// C/D layout: VGPR v, lanes 0-15 -> M=v, lanes 16-31 -> M=v+8, N=lane&15
// ---------------------------------------------------------------------------
__device__ inline v8f wmma_f32(v2f a, v2f b, v8f c) {
    return __builtin_amdgcn_wmma_f32_16x16x4_f32(false, a, false, b, (short)0, c,
                                                 false, false);
}

// ---------------------------------------------------------------------------
// Kernel 0: zero the scatter-max table + compaction counter in workspace
// ---------------------------------------------------------------------------
__global__ void init_kernel(int* ws, int n) {
    int i = blockIdx.x * blockDim.x + threadIdx.x;
    if (i < n) ws[i] = 0;
}

// ---------------------------------------------------------------------------
// Kernel 1: detector. Block = 32 edges, 256 threads (8 waves).
// GEMM (32 x 256) @ (256 x 256) via f32 WMMA: each wave owns 2 N-tiles x 2
// M-subtiles (B fragments reused across M in registers). Then a fully
// parallel Wd2 matvec + sigmoid, then scatter-max of (edge_pos+1).
// ---------------------------------------------------------------------------
__global__ __launch_bounds__(256) void det_kernel(
        const float* __restrict__ emb, const int* __restrict__ eidx,
        const float* __restrict__ Wd1, const float* __restrict__ bd1,
        const float* __restrict__ Wd2, const float* __restrict__ bd2,
        float* __restrict__ scores, int* __restrict__ last)
{
    __shared__ float pair_lds[MT * PSTR];
    __shared__ float h_lds[MT * PSTR];

    const int t    = threadIdx.x;
    const int lane = t & 31;
    const int wave = t >> 5;
    const int e0   = blockIdx.x * MT;
    const int* srcp = eidx;
    const int* dstp = eidx + EE;

    if (t == 0) {  // warm the weight path (global_prefetch_b8)
        __builtin_prefetch(Wd1, 0, 1);
        __builtin_prefetch(Wd2, 0, 1);
    }

    // ---- gather pair tile (32 edges x 256 f32) into LDS: 32 floats/thread ----
    {
        const int row  = t >> 3;          // edge within tile (0..31)
        const int col  = (t & 7) * 32;    // 0..224, 32-float chunk
        const int e    = e0 + row;
        const int node = (col < DD) ? srcp[e] : dstp[e];
        const int goff = (col < DD) ? col : col - DD;
        const float4* g = (const float4*)(emb + (size_t)node * DD + goff);
        float4* l = (float4*)(pair_lds + row * PSTR + col);
        #pragma unroll
        for (int j = 0; j < 8; ++j) l[j] = g[j];
    }
    __syncthreads();

    // ---- (32 x 256) @ (256 x 256): wave = 2 N-tiles, 2 M-subtiles ----
    const int row   = lane & 15;
    const int khalf = (lane >> 4) << 1;   // 0 or 2
    const int ncol  = lane & 15;
    const int n0    = wave * 32;

    v8f acc00 = {}, acc01 = {}, acc10 = {}, acc11 = {};
    for (int k0 = 0; k0 < KK; k0 += 4) {
        v2f a0, a1;
        a0.x = pair_lds[row * PSTR + k0 + khalf];
        a0.y = pair_lds[row * PSTR + k0 + khalf + 1];
        a1.x = pair_lds[(row + 16) * PSTR + k0 + khalf];
        a1.y = pair_lds[(row + 16) * PSTR + k0 + khalf + 1];
        const float* wp = Wd1 + (size_t)(k0 + khalf) * HH;
        v2f b0, b1;
        b0.x = wp[n0 + ncol];        b0.y = wp[HH + n0 + ncol];
        b1.x = wp[n0 + 16 + ncol];   b1.y = wp[HH + n0 + 16 + ncol];
        acc00 = wmma_f32(a0, b0, acc00);
        acc01 = wmma_f32(a0, b1, acc01);
        acc10 = wmma_f32(a1, b0, acc10);
        acc11 = wmma_f32(a1, b1, acc11);
    }

    // ---- bias + ReLU -> h_lds (4 output tiles per wave) ----
    const int mhi = (lane >> 4) << 3;     // 0 or 8
    #pragma unroll
    for (int v = 0; v < 8; ++v) {
        const int M  = v + mhi;
        const int Na = n0 + ncol;
        const int Nb = n0 + 16 + ncol;
        const float ba = bd1[Na], bb = bd1[Nb];
        h_lds[M * PSTR + Na]        = fmaxf(acc00[v] + ba, 0.0f);
        h_lds[M * PSTR + Nb]        = fmaxf(acc01[v] + bb, 0.0f);
        h_lds[(M + 16) * PSTR + Na] = fmaxf(acc10[v] + ba, 0.0f);
        h_lds[(M + 16) * PSTR + Nb] = fmaxf(acc11[v] + bb, 0.0f);
    }
    __syncthreads();

    // ---- Wd2 matvec with all 256 threads (pair tile is dead: reuse as scratch)
    float* part = pair_lds;               // 32 edges x 8 partials (stride 9)
    {
        const int e  = t >> 3;            // 0..31
        const int c0 = (t & 7) * 32;      // 32-element chunk
        float p = 0.0f;
        #pragma unroll 8
        for (int j = 0; j < 32; ++j) p += h_lds[e * PSTR + c0 + j] * Wd2[c0 + j];
        part[e * 9 + (t & 7)] = p;
    }
    __syncthreads();
    if (t < MT) {
        float x = bd2[0];
        #pragma unroll
        for (int j = 0; j < 8; ++j) x += part[t * 9 + j];
        const int e = e0 + t;
        scores[e] = 1.0f / (1.0f + __expf(-x));
        if (x > 0.0f) {                   // sigmoid(x) > 0.5 <=> x > 0
            const int ord = e + 1;
            atomicMax(&last[srcp[e]], ord);
            atomicMax(&last[dstp[e]], ord);
        }
    }
}

// ---------------------------------------------------------------------------
// Kernel 2: compact list of touched nodes
// ---------------------------------------------------------------------------
__global__ void compact_kernel(const int* __restrict__ last, int* counter,
                               int* __restrict__ active) {
    int i = blockIdx.x * blockDim.x + threadIdx.x;
    if (i < NN && last[i] > 0) {
        int p = atomicAdd(counter, 1);
        active[p] = i;
    }
}

// ---------------------------------------------------------------------------
// Kernel 3: passthrough copy for untouched nodes (coalesced float4)
// ---------------------------------------------------------------------------
__global__ void copy_kernel(const float* __restrict__ emb,
                            const int* __restrict__ last,
                            float* __restrict__ out) {
    int t = blockIdx.x * blockDim.x + threadIdx.x;   // NN * 32 threads
    int node = t >> 5, j = t & 31;
    if (node < NN && last[node] == 0)
        ((float4*)out)[(size_t)node * 32 + j] = ((const float4*)emb)[(size_t)node * 32 + j];
}

// ---------------------------------------------------------------------------
// Kernel 4: resolver for touched nodes only. Block = 32 active nodes.
// (32 x 256)@(256 x 256) ReLU then (32 x 256)@(256 x 128), fused epilogue.
// ---------------------------------------------------------------------------
__global__ __launch_bounds__(256) void res_kernel(
        const float* __restrict__ emb, const int* __restrict__ eidx,
        const float* __restrict__ Wr1, const float* __restrict__ br1,
        const float* __restrict__ Wr2, const float* __restrict__ br2,
        const int* __restrict__ last, const int* __restrict__ counter,
        const int* __restrict__ active, float* __restrict__ out)
{
    __shared__ float pair_lds[MT * PSTR];
    __shared__ float h_lds[MT * PSTR];

    const int count = *counter;
    const int base  = blockIdx.x * MT;
    if (base >= count) return;            // uniform early-out, before any barrier

    const int t    = threadIdx.x;
    const int lane = t & 31;
    const int wave = t >> 5;
    const int* srcp = eidx;
    const int* dstp = eidx + EE;

    // ---- gather winning-edge pair per active node ----
    {
        const int row  = t >> 3;
        const int col  = (t & 7) * 32;
        int gi = base + row; if (gi > count - 1) gi = count - 1;  // clamp tail rows
        const int node = active[gi];
        const int e    = last[node] - 1;
        const int nn   = (col < DD) ? srcp[e] : dstp[e];
        const int goff = (col < DD) ? col : col - DD;
        const float4* g = (const float4*)(emb + (size_t)nn * DD + goff);
        float4* l = (float4*)(pair_lds + row * PSTR + col);
        #pragma unroll
        for (int j = 0; j < 8; ++j) l[j] = g[j];
    }
    __syncthreads();

    const int row   = lane & 15;
    const int khalf = (lane >> 4) << 1;
    const int ncol  = lane & 15;
    const int mhi   = (lane >> 4) << 3;

    // ---- GEMM 1: pair @ Wr1, ReLU -> h_lds ----
    {
        const int n0 = wave * 32;
        v8f acc00 = {}, acc01 = {}, acc10 = {}, acc11 = {};
        for (int k0 = 0; k0 < KK; k0 += 4) {
            v2f a0, a1;
            a0.x = pair_lds[row * PSTR + k0 + khalf];
            a0.y = pair_lds[row * PSTR + k0 + khalf + 1];
            a1.x = pair_lds[(row + 16) * PSTR + k0 + khalf];
            a1.y = pair_lds[(row + 16) * PSTR + k0 + khalf + 1];
            const float* wp = Wr1 + (size_t)(k0 + khalf) * HH;
            v2f b0, b1;
            b0.x = wp[n0 + ncol];       b0.y = wp[HH + n0 + ncol];
            b1.x = wp[n0 + 16 + ncol];  b1.y = wp[HH + n0 + 16 + ncol];
            acc00 = wmma_f32(a0, b0, acc00);
            acc01 = wmma_f32(a0, b1, acc01);
            acc10 = wmma_f32(a1, b0, acc10);
            acc11 = wmma_f32(a1, b1, acc11);
        }
        #pragma unroll
        for (int v = 0; v < 8; ++v) {
            const int M  = v + mhi;
            const int Na = n0 + ncol;
            const int Nb = n0 + 16 + ncol;
            const float ba = br1[Na], bb = br1[Nb];
            h_lds[M * PSTR + Na]        = fmaxf(acc00[v] + ba, 0.0f);
            h_lds[M * PSTR + Nb]        = fmaxf(acc01[v] + bb, 0.0f);
            h_lds[(M + 16) * PSTR + Na] = fmaxf(acc10[v] + ba, 0.0f);
            h_lds[(M + 16) * PSTR + Nb] = fmaxf(acc11[v] + bb, 0.0f);
        }
    }
    __syncthreads();

    // ---- GEMM 2: h @ Wr2 (256 x 128): wave = 1 N-tile, 2 M-subtiles ----
    {
        const int n0 = wave * 16;         // 8 waves * 16 = 128 = D
        v8f acc0 = {}, acc1 = {};
        for (int k0 = 0; k0 < KK; k0 += 4) {
            v2f a0, a1;
            a0.x = h_lds[row * PSTR + k0 + khalf];
            a0.y = h_lds[row * PSTR + k0 + khalf + 1];
            a1.x = h_lds[(row + 16) * PSTR + k0 + khalf];
            a1.y = h_lds[(row + 16) * PSTR + k0 + khalf + 1];
            const float* wp = Wr2 + (size_t)(k0 + khalf) * DD;
            v2f b;
            b.x = wp[n0 + ncol];  b.y = wp[DD + n0 + ncol];
            acc0 = wmma_f32(a0, b, acc0);
            acc1 = wmma_f32(a1, b, acc1);
        }
        // fused epilogue: out = (emb + res + br2) * 0.5  (only valid rows)
        #pragma unroll
        for (int v = 0; v < 8; ++v) {
            const int Nc = n0 + ncol;
            const int M  = v + mhi;
            int gi = base + M;
            if (gi < count) {
                const int node = active[gi];
                out[(size_t)node * DD + Nc] =
                    (emb[(size_t)node * DD + Nc] + acc0[v] + br2[Nc]) * 0.5f;
            }
            gi = base + M + 16;
            if (gi < count) {
                const int node = active[gi];
                out[(size_t)node * DD + Nc] =
                    (emb[(size_t)node * DD + Nc] + acc1[v] + br2[Nc]) * 0.5f;
            }
        }
    }
}

// ---------------------------------------------------------------------------
extern "C" void kernel_launch(void* const* d_in, const int* in_sizes, int n_in,
                              void* d_out, int out_size, void* d_ws, size_t ws_size,
                              hipStream_t stream) {
    const float* emb = (const float*)d_in[0];
    const int*   eix = (const int*)  d_in[1];
    const float* Wd1 = (const float*)d_in[2];
    const float* bd1 = (const float*)d_in[3];
    const float* Wd2 = (const float*)d_in[4];
    const float* bd2 = (const float*)d_in[5];
    const float* Wr1 = (const float*)d_in[6];
    const float* br1 = (const float*)d_in[7];
    const float* Wr2 = (const float*)d_in[8];
    const float* br2 = (const float*)d_in[9];

    float* out_emb    = (float*)d_out;                 // (N, D) resolved
    float* out_scores = out_emb + (size_t)NN * DD;     // (E,) scores

    int* ws      = (int*)d_ws;
    int* counter = ws;            // [0]
    int* last    = ws + 16;       // [16 .. 16+N)
    int* activeL = ws + 16 + NN;  // [16+N .. 16+2N)

    const int zn = NN + 16;
    init_kernel<<<(zn + 255) / 256, 256, 0, stream>>>(ws, zn);

    det_kernel<<<EE / MT, 256, 0, stream>>>(emb, eix, Wd1, bd1, Wd2, bd2,
                                            out_scores, last);

    compact_kernel<<<(NN + 255) / 256, 256, 0, stream>>>(last, counter, activeL);

    copy_kernel<<<(NN * 32 + 255) / 256, 256, 0, stream>>>(emb, last, out_emb);

    res_kernel<<<(NN + MT - 1) / MT, 256, 0, stream>>>(emb, eix, Wr1, br1, Wr2, br2,
                                                       last, counter, activeL, out_emb);
}